// HeteroConv_51591147160275
// MI455X (gfx1250) — compile-verified
//
#include <hip/hip_runtime.h>
#include <math.h>

#define N_NODES 50000
#define RTYPES  3
#define NEDGES  600000
#define DIM     128
#define NLAYERS 3
#define BN_EPS  1e-5f
#define SLOPE   0.2f

typedef __attribute__((ext_vector_type(2))) float v2f;
typedef __attribute__((ext_vector_type(8))) float v8f;

// ---------- helpers ----------
__device__ __forceinline__ void atomicMaxF(float* addr, float val) {
    int* ai = (int*)addr;
    int old = *ai;
    while (__int_as_float(old) < val) {
        int prev = atomicCAS(ai, old, __float_as_int(val));
        if (prev == old) break;
        old = prev;
    }
}

// ---------- 1) z = h @ W  via native fp32 WMMA 16x16x4 ----------
// One wave per 16x16 output tile; K=128 in 32 steps of 4.
// A-frag (16x4 f32, 2 VGPR): lanes 0-15 -> K pair {k,k+1}, lanes 16-31 -> {k+2,k+3}
// B-frag (4x16 f32, 2 VGPR): mirrored layout, N striped across lanes.
__global__ void gemm_f32_wmma(const float* __restrict__ A,
                              const float* __restrict__ W,
                              float* __restrict__ Z) {
    int wave = threadIdx.x >> 5;               // 8 waves / block
    int lane = threadIdx.x & 31;
    int tile = blockIdx.x * 8 + wave;          // 25000 tiles total (N%16==0)
    int m0 = (tile >> 3) << 4;                 // row tile (3125 of them)
    int n0 = (tile & 7) << 4;                  // col tile (8 of them)
    int row = lane & 15;
    int kh  = lane >> 4;                       // 0 or 1: which K-half this lane holds

    const float* arow = A + (size_t)(m0 + row) * DIM + 2 * kh;
    const float* bcol = W + (size_t)(2 * kh) * DIM + n0 + row;

    v8f acc = {};
#pragma unroll
    for (int k = 0; k < DIM; k += 4) {
        v2f a = *(const v2f*)(arow + k);       // 8B-aligned pair load
        v2f b;
        b.x = bcol[(size_t)k * DIM];
        b.y = bcol[(size_t)k * DIM + DIM];
        acc = __builtin_amdgcn_wmma_f32_16x16x4_f32(
            /*neg_a=*/false, a, /*neg_b=*/false, b,
            /*c_mod=*/(short)0, acc, /*reuse_a=*/false, /*reuse_b=*/false);
    }
    // C/D layout: VGPR g holds M=g (lanes 0-15) and M=g+8 (lanes 16-31), N=lane%16
    float* zb = Z + (size_t)(m0 + 8 * kh) * DIM + n0 + row;
#pragma unroll
    for (int g = 0; g < 8; ++g)
        zb[(size_t)g * DIM] = acc[g];
}

// ---------- 2) el = z.al, er = z.ar (one wave per node) ----------
__global__ void attn_logits(const float* __restrict__ Z,
                            const float* __restrict__ al,
                            const float* __restrict__ ar,
                            float* __restrict__ el, float* __restrict__ er) {
    int wave = threadIdx.x >> 5;
    int lane = threadIdx.x & 31;
    int node = blockIdx.x * 8 + wave;
    if (node >= N_NODES) return;
    const float* zr = Z + (size_t)node * DIM;
    float sl = 0.f, sr = 0.f;
#pragma unroll
    for (int j = 0; j < 4; ++j) {
        int d = lane + 32 * j;
        float z = zr[d];
        sl += z * al[d];
        sr += z * ar[d];
    }
#pragma unroll
    for (int off = 16; off; off >>= 1) {
        sl += __shfl_xor(sl, off, 32);
        sr += __shfl_xor(sr, off, 32);
    }
    if (lane == 0) { el[node] = sl; er[node] = sr; }
}

// ---------- 3) zero agg, init segment max / sum ----------
__global__ void init_seg(float* __restrict__ agg, float* __restrict__ m,
                         float* __restrict__ ssum) {
    int idx = blockIdx.x * blockDim.x + threadIdx.x;
    if (idx < N_NODES * DIM) agg[idx] = 0.f;
    if (idx < N_NODES) { m[idx] = -INFINITY; ssum[idx] = 0.f; }
}

// ---------- 4) edge logits + leaky relu + segment max ----------
__global__ void edge_logit_max(const int* __restrict__ src, const int* __restrict__ dst,
                               const float* __restrict__ el, const float* __restrict__ er,
                               float* __restrict__ ebuf, float* __restrict__ m) {
    int e = blockIdx.x * blockDim.x + threadIdx.x;
    if (e >= NEDGES) return;
    int d = dst[e];
    float v = el[src[e]] + er[d];
    v = v > 0.f ? v : SLOPE * v;
    ebuf[e] = v;
    atomicMaxF(&m[d], v);
}

// ---------- 5) segment sum of exp(e - m[dst]) ----------
__global__ void edge_expsum(const int* __restrict__ dst, const float* __restrict__ ebuf,
                            const float* __restrict__ m, float* __restrict__ ssum) {
    int e = blockIdx.x * blockDim.x + threadIdx.x;
    if (e >= NEDGES) return;
    int d = dst[e];
    atomicAdd(&ssum[d], __expf(ebuf[e] - m[d]));
}

// ---------- 6) agg[dst] += alpha * z[src] (one wave per edge) ----------
__global__ void edge_scatter(const int* __restrict__ src, const int* __restrict__ dst,
                             const float* __restrict__ ebuf, const float* __restrict__ m,
                             const float* __restrict__ ssum, const float* __restrict__ Z,
                             float* __restrict__ agg) {
    int gid = blockIdx.x * blockDim.x + threadIdx.x;
    int e = gid >> 5;
    int lane = gid & 31;
    if (e >= NEDGES) return;
    int s = src[e], d = dst[e];
    float alpha = __expf(ebuf[e] - m[d]) / ssum[d];
    float4 z = ((const float4*)(Z + (size_t)s * DIM))[lane];
    float* ap = agg + (size_t)d * DIM + lane * 4;
    atomicAdd(ap + 0, alpha * z.x);
    atomicAdd(ap + 1, alpha * z.y);
    atomicAdd(ap + 2, alpha * z.z);
    atomicAdd(ap + 3, alpha * z.w);
}

// ---------- 7) ho (+)= maybe_relu(agg + h + bias)  (sum over relations) ----------
__global__ void combine(const float* __restrict__ agg, const float* __restrict__ h,
                        const float* __restrict__ bias, float* __restrict__ ho,
                        int first, int do_relu) {
    int idx = blockIdx.x * blockDim.x + threadIdx.x;
    if (idx >= N_NODES * DIM) return;
    int d = idx & (DIM - 1);
    float t = agg[idx] + h[idx] + bias[d];
    if (do_relu) t = fmaxf(t, 0.f);
    ho[idx] = first ? t : ho[idx] + t;
}

// ---------- 8) BatchNorm (training-mode, biased var) ----------
__global__ void bn_zero(float* __restrict__ s, float* __restrict__ q) {
    int i = threadIdx.x;
    if (i < DIM) { s[i] = 0.f; q[i] = 0.f; }
}

__global__ void bn_stats(const float* __restrict__ X, float* __restrict__ s,
                         float* __restrict__ q) {
    // 128 blocks, each handles 391 rows; 256 threads = 2 rows x 128 cols
    int col  = threadIdx.x & (DIM - 1);
    int rsub = threadIdx.x >> 7;
    int r0 = blockIdx.x * 391;
    int r1 = r0 + 391; if (r1 > N_NODES) r1 = N_NODES;
    float sum = 0.f, sq = 0.f;
    for (int r = r0 + rsub; r < r1; r += 2) {
        float v = X[(size_t)r * DIM + col];
        sum += v; sq += v * v;
    }
    __shared__ float ls[DIM], lq[DIM];
    if (rsub == 0) { ls[col] = sum; lq[col] = sq; }
    __syncthreads();
    if (rsub == 1) {
        atomicAdd(&s[col], ls[col] + sum);
        atomicAdd(&q[col], lq[col] + sq);
    }
}

__global__ void bn_apply(float* __restrict__ X, const float* __restrict__ s,
                         const float* __restrict__ q, const float* __restrict__ gamma,
                         const float* __restrict__ beta) {
    int idx = blockIdx.x * blockDim.x + threadIdx.x;
    if (idx >= N_NODES * DIM) return;
    int d = idx & (DIM - 1);
    const float inv_n = 1.f / (float)N_NODES;
    float mu  = s[d] * inv_n;
    float var = q[d] * inv_n - mu * mu;
    X[idx] = (X[idx] - mu) * rsqrtf(var + BN_EPS) * gamma[d] + beta[d];
}

// ---------- driver ----------
extern "C" void kernel_launch(void* const* d_in, const int* in_sizes, int n_in,
                              void* d_out, int out_size, void* d_ws, size_t ws_size,
                              hipStream_t stream) {
    const float* x     = (const float*)d_in[0];
    const int*   srcA  = (const int*)d_in[1];
    const int*   dstA  = (const int*)d_in[2];
    const float* Ws    = (const float*)d_in[3];
    const float* al    = (const float*)d_in[4];
    const float* ar    = (const float*)d_in[5];
    const float* bias  = (const float*)d_in[6];
    const float* gamma = (const float*)d_in[7];
    const float* beta  = (const float*)d_in[8];
    float* out = (float*)d_out;

    const size_t ND = (size_t)N_NODES * DIM;
    float* ws  = (float*)d_ws;
    float* hA   = ws; ws += ND;
    float* hB   = ws; ws += ND;
    float* z    = ws; ws += ND;
    float* agg  = ws; ws += ND;
    float* el   = ws; ws += N_NODES;
    float* er   = ws; ws += N_NODES;
    float* mseg = ws; ws += N_NODES;
    float* ssum = ws; ws += N_NODES;
    float* ebuf = ws; ws += NEDGES;
    float* bns  = ws; ws += DIM;
    float* bnq  = ws; ws += DIM;

    const float* lin[NLAYERS]  = { x, hA, hB };
    float*       lout[NLAYERS] = { hA, hB, out };

    const dim3 blk(256);
    const int nd_blocks   = (int)((ND + 255) / 256);          // 25000
    const int edge_blocks = (NEDGES + 255) / 256;             // 2344
    const int scat_blocks = (int)(((size_t)NEDGES * 32 + 255) / 256); // 75000

    for (int i = 0; i < NLAYERS; ++i) {
        const float* h  = lin[i];
        float*       ho = lout[i];
        for (int r = 0; r < RTYPES; ++r) {
            int pr = i * RTYPES + r;
            const int* src = srcA + (size_t)r * NEDGES;
            const int* dst = dstA + (size_t)r * NEDGES;

            gemm_f32_wmma<<<3125, blk, 0, stream>>>(h, Ws + (size_t)pr * DIM * DIM, z);
            attn_logits<<<6250, blk, 0, stream>>>(z, al + (size_t)pr * DIM,
                                                  ar + (size_t)pr * DIM, el, er);
            init_seg<<<nd_blocks, blk, 0, stream>>>(agg, mseg, ssum);
            edge_logit_max<<<edge_blocks, blk, 0, stream>>>(src, dst, el, er, ebuf, mseg);
            edge_expsum<<<edge_blocks, blk, 0, stream>>>(dst, ebuf, mseg, ssum);
            edge_scatter<<<scat_blocks, blk, 0, stream>>>(src, dst, ebuf, mseg, ssum, z, agg);
            combine<<<nd_blocks, blk, 0, stream>>>(agg, h, bias + (size_t)pr * DIM, ho,
                                                   r == 0 ? 1 : 0,
                                                   i < NLAYERS - 1 ? 1 : 0);
        }
        bn_zero<<<1, 128, 0, stream>>>(bns, bnq);
        bn_stats<<<128, blk, 0, stream>>>(ho, bns, bnq);
        bn_apply<<<nd_blocks, blk, 0, stream>>>(ho, bns, bnq,
                                                gamma + (size_t)i * DIM,
                                                beta + (size_t)i * DIM);
    }
}